// Quantization_26053271617878
// MI455X (gfx1250) — compile-verified
//
#include <hip/hip_runtime.h>
#include <hip/hip_bf16.h>

// ---------------------------------------------------------------------------
// Problem constants (from reference): B=16, S=1024 -> M=16384 rows
// IN_DIM = 4864 (K), G*V = 640 (N), G=2, V=320, codevector dim per group = 128
// out = (16384, 256) fp32, then 1 perplexity float appended.
// ---------------------------------------------------------------------------
#define GM   16384
#define GK   4864
#define GN   640
#define NV   320
#define CVD  128

#define ASTRIDE 36   // A LDS row stride (floats): 16B-multiple, conflict-free
#define BSTRIDE 80   // B LDS row stride (floats): 16B-multiple, hi/lo lanes disjoint

typedef __attribute__((ext_vector_type(2))) float v2f;
typedef __attribute__((ext_vector_type(8))) float v8f;

// CDNA5 async memory->LDS copy (ASYNCcnt-tracked), GV addressing, 16B/lane.
__device__ __forceinline__ void async_copy16(uint32_t lds_addr, const float* gaddr) {
    asm volatile("global_load_async_to_lds_b128 %0, %1, off"
                 :: "v"(lds_addr), "v"(gaddr) : "memory");
}

template <int N>
__device__ __forceinline__ void wait_asynccnt() {
#if __has_builtin(__builtin_amdgcn_s_wait_asynccnt)
    __builtin_amdgcn_s_wait_asynccnt(N);
#else
    asm volatile("s_wait_asynccnt %0" :: "i"(N) : "memory");
#endif
}

// ---------------------------------------------------------------------------
// Kernel 1: logits = hs @ W + b  via V_WMMA_F32_16X16X4_F32 (exact fp32 MMA --
// required: argmax over 320 near-N(0,1) logits cannot tolerate bf16 rounding).
// Block tile 128(M) x 64(N), 256 threads = 8 waves; each wave owns a 2x2 grid
// of 16x16 subtiles (32 accum VGPRs) -> 1.0 LDS frag-loads per WMMA.
// Global->LDS staging uses async b128 copies, two LDS buffers; iteration i writes chunk i+1
// into buffer i&1, the same buffer the matrix pipe reads in iteration i.
// ---------------------------------------------------------------------------
__global__ __launch_bounds__(256) void gemm_logits_kernel(
    const float* __restrict__ A,      // [GM][GK]
    const float* __restrict__ Wm,     // [GK][GN]
    const float* __restrict__ bias,   // [GN]
    float* __restrict__ logits)       // [GM][GN]
{
    __shared__ float As[2][128 * ASTRIDE];   // 2 x 18 KB
    __shared__ float Bs[2][32 * BSTRIDE];    // 2 x 10 KB

    const int tid  = threadIdx.x;
    const int lane = tid & 31;
    const int wave = tid >> 5;
    const int hi   = lane >> 4;      // 0: lanes 0-15, 1: lanes 16-31
    const int l15  = lane & 15;

    const int m0 = blockIdx.y * 128;
    const int n0 = blockIdx.x * 64;

    const int s0 = (wave & 3) * 2;        // M subtile pair: strips s0, s0+1 (of 8)
    const int c0 = (wave >> 2) * 2;       // N subtile pair: cols  c0, c0+1 (of 4)

    v8f acc00 = {}, acc01 = {}, acc10 = {}, acc11 = {};

    const uint32_t aLds0 = (uint32_t)(size_t)&As[0][0];
    const uint32_t aLds1 = (uint32_t)(size_t)&As[1][0];
    const uint32_t bLds0 = (uint32_t)(size_t)&Bs[0][0];
    const uint32_t bLds1 = (uint32_t)(size_t)&Bs[1][0];

    // Per-thread async-copy coordinates.
    // A tile 128x32 floats: thread covers 16 consecutive floats of row tid>>1.
    const int ar = tid >> 1;
    const int ac = (tid & 1) * 16;
    // B tile 32x64 floats: thread covers 8 consecutive floats of row tid>>3.
    const int br = tid >> 3;
    const int bc = (tid & 7) * 8;

    const float* agBase = A + (size_t)(m0 + ar) * GK + ac;
    const float* bgBase = Wm + (size_t)br * GN + n0 + bc;
    const uint32_t alOff = (uint32_t)(ar * (ASTRIDE * 4) + ac * 4);
    const uint32_t blOff = (uint32_t)(br * (BSTRIDE * 4) + bc * 4);

    // Prologue: issue chunk 0 into buffer 0 (6 async b128 per thread).
    {
        const float* ag = agBase;               // kc = 0
        const float* bg = bgBase;
        #pragma unroll
        for (int j = 0; j < 4; ++j) async_copy16(aLds0 + alOff + j * 16, ag + j * 4);
        #pragma unroll
        for (int j = 0; j < 2; ++j) async_copy16(bLds0 + blOff + j * 16, bg + j * 4);
    }

    const int NITER = GK / 32;                  // 152 chunks of K=32
    for (int it = 0; it < NITER; ++it) {
        const int cur = it & 1;
        const bool more = (it + 1) < NITER;
        if (more) {
            // Issue next chunk into buffer cur, which is also read below;
            // no barrier orders these writes after this iteration's reads.
            const int kn = (it + 1) * 32;
            const uint32_t aL = (cur ? aLds1 : aLds0) + alOff;
            const uint32_t bL = (cur ? bLds1 : bLds0) + blOff;
            const float* ag = agBase + kn;
            const float* bg = bgBase + (size_t)kn * GN;
            #pragma unroll
            for (int j = 0; j < 4; ++j) async_copy16(aL + j * 16, ag + j * 4);
            #pragma unroll
            for (int j = 0; j < 2; ++j) async_copy16(bL + j * 16, bg + j * 4);
            // In-order completion: <=6 outstanding means the previously issued chunk landed.
            wait_asynccnt<6>();
        } else {
            wait_asynccnt<0>();
        }
        __syncthreads();

        const float* __restrict__ Ab = As[cur];
        const float* __restrict__ Bb = Bs[cur];

        #pragma unroll
        for (int ks = 0; ks < 8; ++ks) {
            const int kk = ks * 4 + hi * 2;
            // A fragments (16x4 MxK): lane m=l15; VGPR0/1 = K {kk,kk+1}; hi +2
            v2f a0, a1;
            a0.x = Ab[((s0 + 0) * 16 + l15) * ASTRIDE + kk];
            a0.y = Ab[((s0 + 0) * 16 + l15) * ASTRIDE + kk + 1];
            a1.x = Ab[((s0 + 1) * 16 + l15) * ASTRIDE + kk];
            a1.y = Ab[((s0 + 1) * 16 + l15) * ASTRIDE + kk + 1];
            // B fragments (4x16 KxN): lane n=l15; VGPR0/1 = K {kk,kk+1}; hi +2
            v2f b0, b1;
            b0.x = Bb[kk * BSTRIDE + (c0 + 0) * 16 + l15];
            b0.y = Bb[(kk + 1) * BSTRIDE + (c0 + 0) * 16 + l15];
            b1.x = Bb[kk * BSTRIDE + (c0 + 1) * 16 + l15];
            b1.y = Bb[(kk + 1) * BSTRIDE + (c0 + 1) * 16 + l15];

            acc00 = __builtin_amdgcn_wmma_f32_16x16x4_f32(false, a0, false, b0, (short)0, acc00, false, false);
            acc01 = __builtin_amdgcn_wmma_f32_16x16x4_f32(false, a0, false, b1, (short)0, acc01, false, false);
            acc10 = __builtin_amdgcn_wmma_f32_16x16x4_f32(false, a1, false, b0, (short)0, acc10, false, false);
            acc11 = __builtin_amdgcn_wmma_f32_16x16x4_f32(false, a1, false, b1, (short)0, acc11, false, false);
        }
        __syncthreads();   // all waves done reading before next async overwrite
    }

    // Epilogue: D layout (16x16 f32): M = v + 8*hi, N = l15
    const int nA = n0 + (c0 + 0) * 16 + l15;
    const int nB = n0 + (c0 + 1) * 16 + l15;
    const float biasA = bias[nA];
    const float biasB = bias[nB];
    #pragma unroll
    for (int v = 0; v < 8; ++v) {
        const int mi0 = m0 + (s0 + 0) * 16 + hi * 8 + v;
        const int mi1 = m0 + (s0 + 1) * 16 + hi * 8 + v;
        logits[(size_t)mi0 * GN + nA] = acc00[v] + biasA;
        logits[(size_t)mi0 * GN + nB] = acc01[v] + biasB;
        logits[(size_t)mi1 * GN + nA] = acc10[v] + biasA;
        logits[(size_t)mi1 * GN + nB] = acc11[v] + biasB;
    }
}

// ---------------------------------------------------------------------------
// Kernel 2: per-row argmax per group (first-max tiebreak, matches jnp.argmax),
// histogram counts, codevector gather. One wave (32 lanes) per row.
// ---------------------------------------------------------------------------
__global__ __launch_bounds__(256) void argmax_gather_kernel(
    const float* __restrict__ logits,  // [GM][GN]
    const float* __restrict__ cv,      // [2][NV][CVD]
    float* __restrict__ out,           // [GM][256]
    int* __restrict__ counts)          // [2*NV]
{
    const int lane = threadIdx.x & 31;
    const int wave = threadIdx.x >> 5;
    const int row  = blockIdx.x * 8 + wave;
    const float* lrow = logits + (size_t)row * GN;

    #pragma unroll
    for (int g = 0; g < 2; ++g) {
        float best = -__builtin_inff();
        int   bidx = 0x7fffffff;
        #pragma unroll
        for (int it = 0; it < NV / 32; ++it) {
            const int v = it * 32 + lane;
            const float x = lrow[g * NV + v];
            if (x > best || (x == best && v < bidx)) { best = x; bidx = v; }
        }
        #pragma unroll
        for (int off = 16; off >= 1; off >>= 1) {
            const float ov = __shfl_xor(best, off, 32);
            const int   oi = __shfl_xor(bidx, off, 32);
            if (ov > best || (ov == best && oi < bidx)) { best = ov; bidx = oi; }
        }
        // all lanes hold the winner after the full xor-butterfly
        if (lane == 0) atomicAdd(&counts[g * NV + bidx], 1);
        const float4 c = *(const float4*)(cv + ((size_t)g * NV + bidx) * CVD + lane * 4);
        *(float4*)(out + (size_t)row * 256 + g * CVD + lane * 4) = c;
    }
}

// ---------------------------------------------------------------------------
// Kernel 3: perplexity = sum_g exp(-sum_v m*log(m+1e-7)), m = count/16384.
// Deterministic single-block LDS tree reduction.
// ---------------------------------------------------------------------------
__global__ __launch_bounds__(512) void perplexity_kernel(
    const int* __restrict__ counts, float* __restrict__ out_ppl)
{
    __shared__ float s0[512];
    __shared__ float s1[512];
    const int t = threadIdx.x;
    float a0 = 0.0f, a1 = 0.0f;
    if (t < NV) {
        const float m0 = (float)counts[t]      * (1.0f / (float)GM);
        const float m1 = (float)counts[NV + t] * (1.0f / (float)GM);
        a0 = m0 * logf(m0 + 1e-7f);
        a1 = m1 * logf(m1 + 1e-7f);
    }
    s0[t] = a0; s1[t] = a1;
    __syncthreads();
    for (int off = 256; off >= 1; off >>= 1) {
        if (t < off) { s0[t] += s0[t + off]; s1[t] += s1[t + off]; }
        __syncthreads();
    }
    if (t == 0) out_ppl[0] = expf(-s0[0]) + expf(-s1[0]);
}

// ---------------------------------------------------------------------------
extern "C" void kernel_launch(void* const* d_in, const int* in_sizes, int n_in,
                              void* d_out, int out_size, void* d_ws, size_t ws_size,
                              hipStream_t stream) {
    const float* hs   = (const float*)d_in[0];   // [16,1024,4864]
    const float* Wm   = (const float*)d_in[1];   // [4864,640]
    const float* bias = (const float*)d_in[2];   // [640]
    const float* cv   = (const float*)d_in[3];   // [1,640,128] == [2][320][128]
    float* out = (float*)d_out;                  // 16384*256 floats + 1 ppl

    float* logits = (float*)d_ws;                               // 42 MB
    int*   counts = (int*)((char*)d_ws + (size_t)GM * GN * sizeof(float));

    (void)hipMemsetAsync(counts, 0, 2 * NV * sizeof(int), stream);

    dim3 ggrid(GN / 64, GM / 128);               // (10, 128)
    gemm_logits_kernel<<<ggrid, 256, 0, stream>>>(hs, Wm, bias, logits);

    argmax_gather_kernel<<<GM / 8, 256, 0, stream>>>(logits, cv, out, counts);

    perplexity_kernel<<<1, 512, 0, stream>>>(counts, out + (size_t)GM * 256);
}